// GraphSage_3298534883955
// MI455X (gfx1250) — compile-verified
//
#include <hip/hip_runtime.h>

// GraphSAGE 2-layer forward for MI455X (gfx1250).
// f32 WMMA 16x16x4 path (exact f32 accumulate vs the f32 reference).
// Workload is L2-gather bound (102MB feature table is L2-resident);
// M=32 row tiles halve the W refetch stream (biggest L2 consumer) by
// reusing each W B-fragment across 2 row tiles / 2 accumulator chains.

typedef float v2f __attribute__((ext_vector_type(2)));
typedef float v4f __attribute__((ext_vector_type(4)));
typedef float v8f __attribute__((ext_vector_type(8)));

#define DIMF   128          // feature dim
#define KNB    10           // sampled neighbors
#define LDC    (2*DIMF + 4) // padded LDS row stride (260 floats) -> conflict-free ds reads

template<int RT>  // number of 16-row tiles per workgroup
__global__ __launch_bounds__(256) void sage_layer_f32(
    const float* __restrict__ feats,     // [n_src, 128] source feature table
    const float* __restrict__ W,         // [128, 256] row-major (out, 2*in)
    const int*   __restrict__ self_idx,  // [rows]
    const int*   __restrict__ neigh_idx, // [rows, KNB]
    float*       __restrict__ out,       // [rows, 128]
    int rows)
{
    constexpr int M = RT * 16;
    __shared__ __align__(16) float comb[M][LDC];   // [self | mean(agg)] tile, Mx256 (+pad)
    __shared__ int sidx[M];
    __shared__ int nidx[M][KNB];

    const int tid  = threadIdx.x;
    const int row0 = blockIdx.x * M;
    if (row0 >= rows) return;   // uniform per block

    // ---- stage indices ----
    for (int i = tid; i < M; i += 256) sidx[i] = self_idx[row0 + i];
    for (int i = tid; i < M * KNB; i += 256) {
        int r = i / KNB, k = i % KNB;
        nidx[r][k] = neigh_idx[(size_t)(row0 + r) * KNB + k];
    }
    __syncthreads();

    // ---- gather + mean into LDS combined tile ----
    // M rows x 32 float4-columns = 32*M slots; 2*RT per thread.
    #pragma unroll
    for (int pp = 0; pp < 2 * RT; ++pp) {
        int p  = tid + pp * 256;
        int r  = p >> 5;        // row in tile
        int c4 = p & 31;        // float4 column (0..31)

        // self half: cols [0,128)
        v4f s = *(const v4f*)(feats + (size_t)sidx[r] * DIMF + c4 * 4);
        *(v4f*)&comb[r][c4 * 4] = s;

        // aggregate half: cols [128,256) = mean over KNB neighbor rows
        v4f acc = {0.f, 0.f, 0.f, 0.f};
        #pragma unroll
        for (int k = 0; k < KNB; ++k) {
            acc += *(const v4f*)(feats + (size_t)nidx[r][k] * DIMF + c4 * 4);
        }
        acc *= (1.0f / KNB);
        *(v4f*)&comb[r][DIMF + c4 * 4] = acc;
    }
    __syncthreads();

    // ---- GEMM: per wave, RT x (16x16) D tiles = comb[Mx256] * W^T ----
    const int lane = tid & 31;
    const int wv   = tid >> 5;      // wave id 0..7 -> N-tile (16 output cols each)
    const int l16  = lane & 15;
    const int hv   = lane >> 4;     // half-wave: selects K sub-pair {0,1} vs {2,3}

    // B frag (4x16 f32): lane l16 holds col N; VGPR0/1 = B[2hv][n], B[2hv+1][n]
    // B[k][n] = W[n][k] (W row-major, rows contiguous 256 floats)
    const float* wbase = W + (size_t)(wv * 16 + l16) * (2 * DIMF) + 2 * hv;

    v8f acc[RT];
    const v8f vzero = {0.f,0.f,0.f,0.f,0.f,0.f,0.f,0.f};
    #pragma unroll
    for (int t = 0; t < RT; ++t) acc[t] = vzero;

    #pragma unroll 8
    for (int ks = 0; ks < 64; ++ks) {
        v2f b = *(const v2f*)(wbase + 4 * ks);  // reused across RT row tiles
        #pragma unroll
        for (int t = 0; t < RT; ++t) {
            // A frag (16x4 f32): lane l16 = row M; VGPR0/1 = K = 2hv, 2hv+1
            v2f a = *(const v2f*)&comb[t * 16 + l16][4 * ks + 2 * hv];
            acc[t] = __builtin_amdgcn_wmma_f32_16x16x4_f32(false, a, false, b,
                                                           (short)0, acc[t], false, false);
        }
    }

    // ---- ReLU + store. D layout: VGPR r -> M = r + 8*hv, N = wv*16 + l16 ----
    #pragma unroll
    for (int t = 0; t < RT; ++t) {
        float* orow = out + (size_t)(row0 + t * 16) * DIMF + (size_t)(wv * 16 + l16);
        #pragma unroll
        for (int r = 0; r < 8; ++r) {
            float v = acc[t][r];
            v = v > 0.f ? v : 0.f;
            orow[(size_t)(r + 8 * hv) * DIMF] = v;
        }
    }
}

extern "C" void kernel_launch(void* const* d_in, const int* in_sizes, int n_in,
                              void* d_out, int out_size, void* d_ws, size_t ws_size,
                              hipStream_t stream) {
    const float* raw = (const float*)d_in[0];   // [200000,128]
    const float* W1  = (const float*)d_in[1];   // [128,256]
    const float* W2  = (const float*)d_in[2];   // [128,256]
    const int* l1    = (const int*)d_in[3];     // [N1]
    const int* n0    = (const int*)d_in[4];     // [N1,10]
    const int* mb    = (const int*)d_in[5];     // [B]
    const int* n1    = (const int*)d_in[6];     // [B,10]

    const int N1 = in_sizes[3];                 // 81920
    const int B  = in_sizes[5];                 // 8192

    float* h1 = (float*)d_ws;                   // [N1,128] = ~42 MB scratch

    constexpr int RT = 2;                       // 32 rows per block
    constexpr int M  = RT * 16;

    // Layer 1: h1 = relu([raw[l1] | mean(raw[n0])] @ W1^T)   (2560 blocks)
    sage_layer_f32<RT><<<(N1 + M - 1) / M, 256, 0, stream>>>(
        raw, W1, l1, n0, h1, N1);

    // Layer 2: out = relu([h1[mb] | mean(h1[n1])] @ W2^T)    (256 blocks)
    sage_layer_f32<RT><<<(B + M - 1) / M, 256, 0, stream>>>(
        h1, W2, mb, n1, (float*)d_out, B);
}